// SORDefense_21904333210072
// MI455X (gfx1250) — compile-verified
//
#include <hip/hip_runtime.h>
#include <math.h>

// ---------------------------------------------------------------------------
// SOR defense (statistical outlier removal) for B=16, K=4096, D=3 point clouds
// Fused pairwise-distance + top-3-smallest using V_WMMA_F32_16X16X4_F32.
// ---------------------------------------------------------------------------

typedef __attribute__((ext_vector_type(2))) float v2f;
typedef __attribute__((ext_vector_type(8))) float v8f;

#define KPTS 4096
#define BATCH 16
#define THREADS 256
#define ROWS_PER_BLOCK 128   // 8 waves * 16 rows each
#define ALPHA_C 1.1f

// Branch-free insert of t into ascending triple (a1 <= a2 <= a3), keeping 3 smallest.
__device__ __forceinline__ void insert3(float& a1, float& a2, float& a3, float t) {
    float n1 = fminf(a1, t);
    float t2 = fmaxf(a1, t);
    float n2 = fminf(a2, t2);
    float t3 = fmaxf(a2, t2);
    float n3 = fminf(a3, t3);
    a1 = n1; a2 = n2; a3 = n3;
}

// Phase 1: per-point mean 2-NN squared distance -> value[b][k]
__global__ __launch_bounds__(THREADS)
void sor_knn_kernel(const float* __restrict__ x, float* __restrict__ value) {
    // Padded LDS image of this batch's points: {-2x0, -2x1, -2x2, ||x||^2}
    __shared__ float XB[KPTS][4];

    const int b   = blockIdx.y;
    const int tid = threadIdx.x;
    const float* xb = x + (size_t)b * KPTS * 3;

    // Stage the whole batch into LDS (64 KB), padded form.
    for (int p = tid; p < KPTS; p += THREADS) {
        float x0 = xb[p * 3 + 0];
        float x1 = xb[p * 3 + 1];
        float x2 = xb[p * 3 + 2];
        float xx = x0 * x0 + x1 * x1 + x2 * x2;
        XB[p][0] = -2.0f * x0;
        XB[p][1] = -2.0f * x1;
        XB[p][2] = -2.0f * x2;
        XB[p][3] = xx;
    }
    __syncthreads();

    const int wave = tid >> 5;         // wave32
    const int lane = tid & 31;
    const int half = lane >> 4;        // 0: lanes 0-15, 1: lanes 16-31
    const int l15  = lane & 15;
    const int m0   = blockIdx.x * ROWS_PER_BLOCK + wave * 16;   // row tile base

    // A-matrix 16x4 f32 layout: lane<16 holds {K0,K1}, lane>=16 holds {K2,K3}.
    // Derive A = {x0,x1,x2,1} from the XB image: scale by -0.5, patch K3 -> 1.0.
    v2f a = *(const v2f*)&XB[m0 + l15][half * 2];
    a[0] *= -0.5f;
    a[1] *= -0.5f;
    if (half) a[1] = 1.0f;             // K=3 pad column of A is 1.0

    // C init: C VGPR v holds row M = half*8 + v -> preload ||x_row||^2.
    v8f cinit;
#pragma unroll
    for (int v = 0; v < 8; ++v) cinit[v] = XB[m0 + half * 8 + v][3];

    // Running 3-smallest per accumulator slot (row = half*8+v, cols == l15 mod 16).
    float d1[8], d2[8], d3[8];
#pragma unroll
    for (int v = 0; v < 8; ++v) { d1[v] = 3.0e38f; d2[v] = 3.0e38f; d3[v] = 3.0e38f; }

    // Sweep all 4096 columns in 16-wide tiles: one WMMA per tile yields the
    // full 16x16 squared-distance tile (D = A*B + C).
    for (int n0 = 0; n0 < KPTS; n0 += 16) {
        v2f bmat = *(const v2f*)&XB[n0 + l15][half * 2];
        v8f d = __builtin_amdgcn_wmma_f32_16x16x4_f32(
            /*neg_a=*/false, a, /*neg_b=*/false, bmat,
            /*c_mod=*/(short)0, cinit, /*reuse_a=*/false, /*reuse_b=*/false);
#pragma unroll
        for (int v = 0; v < 8; ++v) insert3(d1[v], d2[v], d3[v], d[v]);
    }

    // Merge triples across the 16 lanes of each half (xor 1,2,4,8 stays in-half).
#pragma unroll
    for (int m = 1; m <= 8; m <<= 1) {
#pragma unroll
        for (int v = 0; v < 8; ++v) {
            float o1 = __shfl_xor(d1[v], m, 32);
            float o2 = __shfl_xor(d2[v], m, 32);
            float o3 = __shfl_xor(d3[v], m, 32);
            insert3(d1[v], d2[v], d3[v], o1);
            insert3(d1[v], d2[v], d3[v], o2);
            insert3(d1[v], d2[v], d3[v], o3);
        }
    }

    // d1 ~ self distance (dropped, like the reference); value = mean(d2, d3).
    if (l15 == 0) {
        float* out = value + (size_t)b * KPTS + m0 + half * 8;
#pragma unroll
        for (int v = 0; v < 8; ++v) out[v] = 0.5f * (d2[v] + d3[v]);
    }
}

// Phase 2: per-batch mean/std(ddof=1) -> threshold -> masked points + mask.
__global__ __launch_bounds__(THREADS)
void sor_mask_kernel(const float* __restrict__ x, const float* __restrict__ value,
                     float* __restrict__ sel_pc, float* __restrict__ mask) {
    const int b   = blockIdx.x;
    const int tid = threadIdx.x;
    const float* vb = value + (size_t)b * KPTS;

    float s = 0.0f, ss = 0.0f;
    for (int i = tid; i < KPTS; i += THREADS) {
        float t = vb[i];
        s  += t;
        ss += t * t;
    }
#pragma unroll
    for (int m = 16; m >= 1; m >>= 1) {
        s  += __shfl_xor(s,  m, 32);
        ss += __shfl_xor(ss, m, 32);
    }
    __shared__ float sb[8], ssb[8];
    __shared__ float thr_s;
    if ((tid & 31) == 0) { sb[tid >> 5] = s; ssb[tid >> 5] = ss; }
    __syncthreads();
    if (tid == 0) {
        float S = 0.0f, SS = 0.0f;
        for (int i = 0; i < 8; ++i) { S += sb[i]; SS += ssb[i]; }
        float mean = S / (float)KPTS;
        float var  = (SS - (float)KPTS * mean * mean) / (float)(KPTS - 1);
        var = var > 0.0f ? var : 0.0f;
        thr_s = mean + ALPHA_C * sqrtf(var);
    }
    __syncthreads();
    float thr = thr_s;

    const float* xb = x + (size_t)b * KPTS * 3;
    float* sp = sel_pc + (size_t)b * KPTS * 3;
    float* mk = mask + (size_t)b * KPTS;
    for (int i = tid; i < KPTS; i += THREADS) {
        float keep = (vb[i] <= thr) ? 1.0f : 0.0f;
        mk[i] = keep;
        sp[i * 3 + 0] = xb[i * 3 + 0] * keep;
        sp[i * 3 + 1] = xb[i * 3 + 1] * keep;
        sp[i * 3 + 2] = xb[i * 3 + 2] * keep;
    }
}

extern "C" void kernel_launch(void* const* d_in, const int* in_sizes, int n_in,
                              void* d_out, int out_size, void* d_ws, size_t ws_size,
                              hipStream_t stream) {
    (void)in_sizes; (void)n_in; (void)out_size; (void)ws_size;
    const float* x = (const float*)d_in[0];
    float* out    = (float*)d_out;
    float* value  = (float*)d_ws;                    // 16*4096 floats = 256 KB

    float* sel_pc = out;                             // [16,4096,3]
    float* mask   = out + (size_t)BATCH * KPTS * 3;  // [16,4096] as 0/1 floats

    dim3 grid1(KPTS / ROWS_PER_BLOCK, BATCH);        // (32, 16)
    sor_knn_kernel<<<grid1, THREADS, 0, stream>>>(x, value);
    sor_mask_kernel<<<BATCH, THREADS, 0, stream>>>(x, value, sel_pc, mask);
}